// HashGridEncode_51539607824
// MI455X (gfx1250) — compile-verified
//
#include <hip/hip_runtime.h>
#include <hip/hip_bf16.h>

namespace {
constexpr int kNumLevels = 16;
constexpr int kOff[kNumLevels + 1] = {
    0, 4096, 12096, 27721, 60489, 124489, 249489, 511633, 1023633,
    1547921, 2072209, 2596497, 3120785, 3645073, 4169361, 4693649, 5217937};
constexpr int kRes[kNumLevels] = {16, 20, 25, 32, 40, 50, 64, 80,
                                  101, 128, 161, 203, 256, 322, 406, 512};
// Levels 8..15 all have table size 524288 = 2^19  ->  idx % size == idx & mask
constexpr unsigned kHashMask = 0x7FFFFu;
// Stage dense levels 0..1 into LDS: 12096 float2 entries = 94.5 KB (< 320 KB/WGP)
constexpr int kLdsEntries = kOff[2];
constexpr int kBlock = 256;
}  // namespace

// Types matching the async-LDS builtin signature probed in round 1:
// (v2i addrspace(1)* src_global, v2i addrspace(3)* dst_lds, imm offset, imm cpol)
typedef int v2i __attribute__((ext_vector_type(2)));
typedef float v4f __attribute__((ext_vector_type(4)));
typedef __attribute__((address_space(1))) v2i* async_gptr_t;
typedef __attribute__((address_space(3))) v2i* async_lptr_t;

#define HAS_ASYNC_LDS __has_builtin(__builtin_amdgcn_global_load_async_to_lds_b64)

__global__ __launch_bounds__(kBlock) void hashgrid_encode_kernel(
    const float* __restrict__ xyz, const float* __restrict__ emb,
    const float* __restrict__ minp, const float* __restrict__ maxp,
    float* __restrict__ out, int npts)
{
    __shared__ float2 s_emb[kLdsEntries];
    const float2* __restrict__ embv = reinterpret_cast<const float2*>(emb);

    // ---- Stage levels 0..1 into LDS via CDNA5 async global->LDS DMA ----
#if HAS_ASYNC_LDS
    for (int i = (int)threadIdx.x; i < kLdsEntries; i += kBlock) {
        __builtin_amdgcn_global_load_async_to_lds_b64(
            (async_gptr_t)(embv + i),
            (async_lptr_t)(&s_emb[i]),
            /*offset=*/0, /*cpol=*/0);
    }
  #if __has_builtin(__builtin_amdgcn_s_wait_asynccnt)
    __builtin_amdgcn_s_wait_asynccnt(0);
  #endif
#else
    for (int i = (int)threadIdx.x; i < kLdsEntries; i += kBlock)
        s_emb[i] = embv[i];
#endif
    __syncthreads();

    const int b = (int)blockIdx.x * kBlock + (int)threadIdx.x;
    if (b >= npts) return;

    // Streaming input: non-temporal so xyz doesn't pollute the L2-resident table
    const float X = __builtin_nontemporal_load(&xyz[3 * b + 0]);
    const float Y = __builtin_nontemporal_load(&xyz[3 * b + 1]);
    const float Z = __builtin_nontemporal_load(&xyz[3 * b + 2]);
    const float mnx = minp[0], mny = minp[1], mnz = minp[2];
    const float xn = (X - mnx) / (maxp[0] - mnx);
    const float yn = (Y - mny) / (maxp[1] - mny);
    const float zn = (Z - mnz) / (maxp[2] - mnz);

    float acc[2 * kNumLevels];

#pragma unroll
    for (int l = 0; l < kNumLevels; ++l) {
        const int   res = kRes[l];
        const float rf  = (float)(res - 1);
        const float hi  = (float)(res - 2);

        const float px = xn * rf, py = yn * rf, pz = zn * rf;
        const float fx = fminf(fmaxf(floorf(px), 0.0f), hi);
        const float fy = fminf(fmaxf(floorf(py), 0.0f), hi);
        const float fz = fminf(fmaxf(floorf(pz), 0.0f), hi);
        const float wx = px - fx, wy = py - fy, wz = pz - fz;
        const unsigned x0 = (unsigned)fx, y0 = (unsigned)fy, z0 = (unsigned)fz;

        // Corner c: cx = (c>>2)&1, cy = (c>>1)&1, cz = c&1  (matches reference CORNERS)
        unsigned idx[8];
        if (l < 8) {
            // dense level: idx = x*res^2 + y*res + z
            const unsigned ru = (unsigned)res;
            const unsigned rr = ru * ru;
            const unsigned base = x0 * rr + y0 * ru + z0;
            idx[0] = base;            idx[1] = base + 1u;
            idx[2] = base + ru;       idx[3] = base + ru + 1u;
            idx[4] = base + rr;       idx[5] = base + rr + 1u;
            idx[6] = base + rr + ru;  idx[7] = base + rr + ru + 1u;
        } else {
            // hashed level: (x*1) ^ (y*p1) ^ (z*p2), uint32 wraparound, size = 2^19
            const unsigned hx0 = x0,                    hx1 = x0 + 1u;
            const unsigned hy0 = y0 * 2654435761u,      hy1 = (y0 + 1u) * 2654435761u;
            const unsigned hz0 = z0 * 805459861u,       hz1 = (z0 + 1u) * 805459861u;
            idx[0] = (hx0 ^ hy0 ^ hz0) & kHashMask;
            idx[1] = (hx0 ^ hy0 ^ hz1) & kHashMask;
            idx[2] = (hx0 ^ hy1 ^ hz0) & kHashMask;
            idx[3] = (hx0 ^ hy1 ^ hz1) & kHashMask;
            idx[4] = (hx1 ^ hy0 ^ hz0) & kHashMask;
            idx[5] = (hx1 ^ hy0 ^ hz1) & kHashMask;
            idx[6] = (hx1 ^ hy1 ^ hz0) & kHashMask;
            idx[7] = (hx1 ^ hy1 ^ hz1) & kHashMask;
        }

        // Gathers: levels 0..1 from LDS (ds_load_2addr_b64 pairs), rest from the
        // L2-resident table with default (RT) temporal hint for reuse.
        float2 f[8];
        if (l < 2) {
#pragma unroll
            for (int c = 0; c < 8; ++c) f[c] = s_emb[idx[c] + (unsigned)kOff[l]];
        } else {
#pragma unroll
            for (int c = 0; c < 8; ++c) f[c] = embv[idx[c] + (unsigned)kOff[l]];
        }

        const float wx0 = 1.0f - wx, wy0 = 1.0f - wy, wz0 = 1.0f - wz;
        const float c00 = wx0 * wy0, c01 = wx0 * wy, c10 = wx * wy0, c11 = wx * wy;
        const float wt[8] = {c00 * wz0, c00 * wz, c01 * wz0, c01 * wz,
                             c10 * wz0, c10 * wz, c11 * wz0, c11 * wz};

        float a0 = 0.0f, a1 = 0.0f;
#pragma unroll
        for (int c = 0; c < 8; ++c) {
            a0 = fmaf(wt[c], f[c].x, a0);
            a1 = fmaf(wt[c], f[c].y, a1);
        }
        acc[2 * l]     = a0;
        acc[2 * l + 1] = a1;
    }

    // Contiguous 128 B per thread -> 8 x b128 non-temporal stores (pure stream,
    // never re-read by the kernel; keep it out of L2 so the table stays hot).
    v4f* __restrict__ o4 = reinterpret_cast<v4f*>(out + (size_t)b * (2 * kNumLevels));
#pragma unroll
    for (int i = 0; i < 8; ++i) {
        v4f v = {acc[4 * i + 0], acc[4 * i + 1], acc[4 * i + 2], acc[4 * i + 3]};
        __builtin_nontemporal_store(v, &o4[i]);
    }
}

extern "C" void kernel_launch(void* const* d_in, const int* in_sizes, int n_in,
                              void* d_out, int out_size, void* d_ws, size_t ws_size,
                              hipStream_t stream) {
    const float* xyz = (const float*)d_in[0];
    const float* emb = (const float*)d_in[1];
    const float* mn  = (const float*)d_in[2];
    const float* mx  = (const float*)d_in[3];
    float* out = (float*)d_out;
    const int npts = in_sizes[0] / 3;
    const int grid = (npts + kBlock - 1) / kBlock;
    hashgrid_encode_kernel<<<grid, kBlock, 0, stream>>>(xyz, emb, mn, mx, out, npts);
}